// Spark_21131239097064
// MI455X (gfx1250) — compile-verified
//
#include <hip/hip_runtime.h>
#include <cfloat>

#define KB   8
#define KN   1024
#define KDX  64
#define KHID 128
#define KKRW 4
#define KD1  65
#define KH1  129
#define KD1P 80     // padded K-stride for 65-dim hyperbolic rows (mult of 16)
#define KH1P 160    // padded stride for 129-dim hyperbolic rows (mult of 32: used as Nc)
#define EPS_     1e-7f
#define MINNORM_ 1e-15f

typedef __attribute__((ext_vector_type(2))) float v2f;
typedef __attribute__((ext_vector_type(8))) float v8f;

// ---------------- device helpers ----------------
__device__ __forceinline__ float d_softplus(float x){ return x > 20.f ? x : log1pf(expf(x)); }
__device__ __forceinline__ float d_mish(float x){ return x * tanhf(d_softplus(x)); }
__device__ __forceinline__ float d_sigmoid(float x){ return 1.f / (1.f + expf(-x)); }
__device__ __forceinline__ float d_arcosh(float x){ return logf(x + sqrtf(fmaxf(x*x - 1.f, 1e-14f))); }

// block-wide sum; blockDim.x must be 256 for all callers
__device__ float blk_sum(float v){
  __shared__ float s[256];
  int t = threadIdx.x;
  __syncthreads();
  s[t] = v; __syncthreads();
  for (int k = 128; k > 0; k >>= 1){ if (t < k) s[t] += s[t + k]; __syncthreads(); }
  return s[0];
}

// ---------------- WMMA GEMM (full tiles: M%64==0, Nc%32==0, K%4==0) ----------------
// C[b](MxNc) = A[b](MxK,row-major) @ B(KxNc)   (transB: B stored NcxK row-major)
// grid = (Nc/32, M/64, batch); block = 128 (4 waves; each wave computes a 16x32 C strip
// as two 16x16 WMMA tiles sharing the A fragment -> 3 loads per 2 WMMAs)
__global__ void k_gemm(float* __restrict__ C, const float* __restrict__ A,
                       const float* __restrict__ Bm,
                       int K, int Nc, long sA, long sB, long sC, int transB)
{
  int b = blockIdx.z;
  const float* Ab = A + (size_t)b * (size_t)sA;
  const float* Bb = Bm + (size_t)b * (size_t)sB;
  float* Cb = C + (size_t)b * (size_t)sC;
  int wave = threadIdx.x >> 5;
  int lane = threadIdx.x & 31;
  int h = lane >> 4, l = lane & 15;
  int row  = blockIdx.y * 64 + wave * 16 + l;
  int col0 = blockIdx.x * 32 + l;
  const float* ap = Ab + (size_t)row * K + 2 * h;   // ISA: A vgpr v, half h -> K = v + 2h
  v8f acc0 = {}, acc1 = {};
  if (transB) {
    const float* bp0 = Bb + (size_t)col0 * K + 2 * h;
    const float* bp1 = bp0 + (size_t)16 * K;
    #pragma unroll 4
    for (int kk = 0; kk < K; kk += 4) {
      v2f av = *(const v2f*)(ap + kk);
      v2f b0 = *(const v2f*)(bp0 + kk);
      v2f b1 = *(const v2f*)(bp1 + kk);
      acc0 = __builtin_amdgcn_wmma_f32_16x16x4_f32(false, av, false, b0, (short)0, acc0, false, false);
      acc1 = __builtin_amdgcn_wmma_f32_16x16x4_f32(false, av, false, b1, (short)0, acc1, false, false);
    }
  } else {
    const float* bp = Bb + (size_t)(2 * h) * Nc + col0;
    #pragma unroll 4
    for (int kk = 0; kk < K; kk += 4) {
      v2f av = *(const v2f*)(ap + kk);
      size_t o = (size_t)kk * Nc;
      v2f b0, b1;
      b0.x = bp[o];      b0.y = bp[o + Nc];
      b1.x = bp[o + 16]; b1.y = bp[o + Nc + 16];
      acc0 = __builtin_amdgcn_wmma_f32_16x16x4_f32(false, av, false, b0, (short)0, acc0, false, false);
      acc1 = __builtin_amdgcn_wmma_f32_16x16x4_f32(false, av, false, b1, (short)0, acc1, false, false);
    }
  }
  int orow = blockIdx.y * 64 + wave * 16 + 8 * h;   // ISA: C vgpr r -> M = r + 8h
  #pragma unroll
  for (int r = 0; r < 8; ++r) {
    size_t off = (size_t)(orow + r) * Nc + col0;
    Cb[off]      = acc0[r];
    Cb[off + 16] = acc1[r];
  }
}

// zero-padded weight copy: dst(Rp x Cp) <- src(R x Cc)
__global__ void k_padW(float* dst, const float* __restrict__ src, int R, int Cc, int Rp, int Cp)
{
  int idx = blockIdx.x * 256 + threadIdx.x;
  if (idx >= Rp * Cp) return;
  int r = idx / Cp, c = idx % Cp;
  dst[idx] = (r < R && c < Cc) ? src[r * Cc + c] : 0.f;
}

// ---------------- RRWP ----------------
__global__ void k_rrwp_T(float* __restrict__ T, const float* __restrict__ adj,
                         const float* __restrict__ w)
{
  int b = blockIdx.y, i = blockIdx.x, t = threadIdx.x;
  size_t ro = ((size_t)b * KN + i) * KN;
  float s = 0.f;
  for (int j = t; j < KN; j += 256) s += adj[ro + j] * fabsf(w[ro + j]);
  float tot = blk_sum(s);
  float inv = 1.f / fmaxf(tot, 1e-8f);
  for (int j = t; j < KN; j += 256) T[ro + j] = adj[ro + j] * fabsf(w[ro + j]) * inv;
}

__global__ void k_diag(float* p, int col, const float* __restrict__ M)
{
  int idx = blockIdx.x * 256 + threadIdx.x;
  if (idx >= KB * KN) return;
  int i = idx % KN;
  p[(size_t)idx * KKRW + col] = M[(size_t)idx * KN + i];
}

__global__ void k_diag4(float* p, const float* __restrict__ A, const float* __restrict__ Bm)
{
  int b = blockIdx.y, i = blockIdx.x, t = threadIdx.x;
  size_t rb = (size_t)b * KN * KN;
  float s = 0.f;
  for (int j = t; j < KN; j += 256)
    s += A[rb + (size_t)i * KN + j] * Bm[rb + (size_t)j * KN + i];
  float tot = blk_sum(s);
  if (t == 0) p[((size_t)b * KN + i) * KKRW + 3] = tot;
}

// ---------------- fuse + layernorm ----------------
__global__ void k_fuse_ln(float* xp, const float* __restrict__ x, const float* __restrict__ p,
                          const float* __restrict__ fw, const float* __restrict__ fb,
                          const float* __restrict__ g, const float* __restrict__ be)
{
  int b = blockIdx.y, i = blockIdx.x, t = threadIdx.x;
  size_t r = (size_t)b * KN + i;
  float v = 0.f;
  if (t < KDX) {
    v = x[r * KDX + t] + fb[t];
    for (int k = 0; k < KKRW; k++) v += p[r * KKRW + k] * fw[t * KKRW + k];
  }
  float s1 = blk_sum(t < KDX ? v : 0.f);
  float s2 = blk_sum(t < KDX ? v * v : 0.f);
  float m = s1 / KDX, var = s2 / KDX - m * m;
  if (t < KDX) xp[r * KDX + t] = (v - m) * rsqrtf(var + 1e-5f) * g[t] + be[t];
}

// ---------------- cheb1 helpers ----------------
__global__ void k_dinv(float* dinv, const float* __restrict__ e)
{
  int b = blockIdx.y, i = blockIdx.x, t = threadIdx.x;
  size_t ro = ((size_t)b * KN + i) * KN;
  float s = 0.f;
  for (int j = t; j < KN; j += 256) s += e[ro + j];
  float tot = blk_sum(s);
  if (t == 0) dinv[(size_t)b * KN + i] = 1.f / sqrtf(fmaxf(tot, 1e-8f));
}

__global__ void k_make_an(float* an, const float* __restrict__ e, const float* __restrict__ dinv)
{
  size_t idx = (size_t)blockIdx.x * 256 + threadIdx.x;
  if (idx >= (size_t)KB * KN * KN) return;
  size_t b = idx / ((size_t)KN * KN);
  size_t rem = idx - b * (size_t)KN * KN;
  int i = (int)(rem / KN), j = (int)(rem % KN);
  an[idx] = e[idx] * dinv[b * KN + i] * dinv[b * KN + j];
}

__global__ void k_combine(float* out, const float* __restrict__ U, const float* __restrict__ V,
                          const float* __restrict__ bias, int D, int n, int domish)
{
  int idx = blockIdx.x * 256 + threadIdx.x;
  if (idx >= n) return;
  float r = U[idx] - V[idx] + bias[idx % D];
  out[idx] = domish ? d_mish(r) : r;
}

// ---------------- hyperbolic row kernels (stride DS >= D; padding kept zero) ----------
__global__ void k_logmap(float* out, const float* __restrict__ in, int D, int DS)
{
  int b = blockIdx.y, i = blockIdx.x, t = threadIdx.x;
  size_t r = ((size_t)b * KN + i) * (size_t)DS;
  float v = (t >= 1 && t < D) ? in[r + t] : 0.f;
  float yn = fmaxf(sqrtf(blk_sum(v * v)), MINNORM_);
  float x0 = fmaxf(in[r], 1.f + EPS_);
  float sc = d_arcosh(x0) / yn;
  if (t < DS) out[r + t] = (t == 0 || t >= D) ? 0.f : v * sc;
}

// proj(expmap0(u)); embed=1: input rows are (D-1) wide (stride DSi), mapped to comps 1..D-1
__global__ void k_expproj(float* out, const float* __restrict__ in, int D, int embed,
                          int DSo, int DSi)
{
  int b = blockIdx.y, i = blockIdx.x, t = threadIdx.x;
  size_t ri = ((size_t)b * KN + i) * (size_t)DSi;
  size_t ro = ((size_t)b * KN + i) * (size_t)DSo;
  float y = 0.f;
  if (t >= 1 && t < D) y = embed ? in[ri + t - 1] : in[ri + t];
  float yn = fmaxf(sqrtf(blk_sum(y * y)), MINNORM_);
  float s = sinhf(yn) / yn;
  float o = s * y;
  float s2 = blk_sum(o * o);
  if (t == 0) out[ro] = sqrtf(1.f + s2);
  else if (t < DSo) out[ro + t] = (t < D) ? o : 0.f;
}

// precompute lhb = logmap0(proj(expmap0(proj_tan0(b))))
__global__ void k_hbprep(float* lhb, const float* __restrict__ bv, int D)
{
  int t = threadIdx.x;
  float u = (t >= 1 && t < D) ? bv[t] : 0.f;
  float yn = fmaxf(sqrtf(blk_sum(u * u)), MINNORM_);
  float s = sinhf(yn) / yn;
  float hb = s * u;
  float sum2 = blk_sum(hb * hb);
  float x0 = sqrtf(1.f + sum2);
  float yn2 = fmaxf(sqrtf(sum2), MINNORM_);
  float ac = d_arcosh(fmaxf(x0, 1.f + EPS_));
  if (t < D) lhb[t] = (t == 0) ? 0.f : ac * hb / yn2;
}

// in-place mobius_add(h, hb) = proj(expmap(proj_tan(lhb - alpha*v, h), h))
__global__ void k_mobius(float* H, const float* __restrict__ lhb, int D, int DS)
{
  int b = blockIdx.y, i = blockIdx.x, t = threadIdx.x;
  size_t r = ((size_t)b * KN + i) * (size_t)DS;
  float x0 = H[r];
  float y = (t >= 1 && t < D) ? H[r + t] : 0.f;
  float yn = fmaxf(sqrtf(blk_sum(y * y)), MINNORM_);
  float yh = y / yn;
  float lb = (t >= 1 && t < D) ? lhb[t] : 0.f;
  float alpha = blk_sum(yh * lb);
  float u = lb - alpha * (1.f - x0) * yh;   // comps >= 1
  float ux = blk_sum(y * u);
  float w0 = ux / fmaxf(x0, EPS_);
  float sw = blk_sum(u * u);
  float dot = sw - w0 * w0;                 // sum(w^2) - 2*w0^2
  float theta = fmaxf(fminf(sqrtf(fmaxf(dot, EPS_)), 1000.f), MINNORM_);
  float ch = coshf(theta), st = sinhf(theta) / theta;
  float o = ch * y + st * u;
  float so = blk_sum(o * o);
  if (t == 0) H[r] = sqrtf(1.f + so);
  else if (t < D) H[r + t] = o;
}

// in-place: h = proj(expmap0(proj_tan0(mish(logmap0(h)))))
__global__ void k_mish_rows(float* H, int D, int DS)
{
  int b = blockIdx.y, i = blockIdx.x, t = threadIdx.x;
  size_t r = ((size_t)b * KN + i) * (size_t)DS;
  float x0 = H[r];
  float y = (t >= 1 && t < D) ? H[r + t] : 0.f;
  float yn = fmaxf(sqrtf(blk_sum(y * y)), MINNORM_);
  float ac = d_arcosh(fmaxf(x0, 1.f + EPS_));
  float m = d_mish(ac * y / yn);
  float mn = fmaxf(sqrtf(blk_sum(m * m)), MINNORM_);
  float s = sinhf(mn) / mn;
  float o = s * m;
  float so = blk_sum(o * o);
  if (t == 0) H[r] = sqrtf(1.f + so);
  else if (t < D) H[r + t] = o;
}

// ---------------- distances ----------------
__global__ void k_diagvec(float* sq, const float* __restrict__ G)
{
  int idx = blockIdx.x * 256 + threadIdx.x;
  if (idx >= KB * KN) return;
  sq[idx] = G[(size_t)idx * KN + (idx % KN)];
}
__global__ void k_col0(float* h0, const float* __restrict__ H, int DS)
{
  int idx = blockIdx.x * 256 + threadIdx.x;
  if (idx >= KB * KN) return;
  h0[idx] = H[(size_t)idx * DS];
}
__global__ void k_dist_e(float* de, const float* __restrict__ G, const float* __restrict__ sq)
{
  size_t idx = (size_t)blockIdx.x * 256 + threadIdx.x;
  if (idx >= (size_t)KB * KN * KN) return;
  size_t b = idx / ((size_t)KN * KN);
  size_t rem = idx - b * (size_t)KN * KN;
  int i = (int)(rem / KN), j = (int)(rem % KN);
  if (i == j) { de[idx] = 0.f; return; }
  size_t base = b * (size_t)KN * KN;
  float si = sq[b * KN + i], sj = sq[b * KN + j];
  float f1 = sqrtf(fmaxf(si + sj - 2.f * G[base + (size_t)i * KN + j], 1e-12f));
  float f2 = sqrtf(fmaxf(si + sj - 2.f * G[base + (size_t)j * KN + i], 1e-12f));
  de[idx] = 0.5f * (f1 + f2);
}
__device__ __forceinline__ float hyp_d(float g, float h0i, float h0j)
{
  float th = fmaxf(-(g - 2.f * h0i * h0j), 1.f + EPS_);
  float ac = d_arcosh(th);
  float sd = fminf(fmaxf(ac * ac, 0.f), 50.f);
  return sqrtf(fmaxf(sd, 1e-12f));
}
__global__ void k_dist_h(float* dh, const float* __restrict__ G, const float* __restrict__ h0)
{
  size_t idx = (size_t)blockIdx.x * 256 + threadIdx.x;
  if (idx >= (size_t)KB * KN * KN) return;
  size_t b = idx / ((size_t)KN * KN);
  size_t rem = idx - b * (size_t)KN * KN;
  int i = (int)(rem / KN), j = (int)(rem % KN);
  if (i == j) { dh[idx] = 0.f; return; }
  size_t base = b * (size_t)KN * KN;
  float hi = h0[b * KN + i], hj = h0[b * KN + j];
  float f1 = hyp_d(G[base + (size_t)i * KN + j], hi, hj);
  float f2 = hyp_d(G[base + (size_t)j * KN + i], hj, hi);
  dh[idx] = 0.5f * (f1 + f2);
}

// ---------------- off-diagonal min/max (two stage) ----------------
__global__ void k_mm1(float* pmin, float* pmax, const float* __restrict__ M)
{
  __shared__ float smn[256], smx[256];
  int b = blockIdx.y, c = blockIdx.x, t = threadIdx.x;
  size_t base = (size_t)b * KN * KN;
  float mn = FLT_MAX, mx = -FLT_MAX;
  for (int it = 0; it < 32; it++) {
    int pos = c * 8192 + it * 256 + t;
    int i = pos >> 10, j = pos & 1023;
    if (i != j) { float v = M[base + pos]; mn = fminf(mn, v); mx = fmaxf(mx, v); }
  }
  smn[t] = mn; smx[t] = mx; __syncthreads();
  for (int k = 128; k > 0; k >>= 1) {
    if (t < k) { smn[t] = fminf(smn[t], smn[t + k]); smx[t] = fmaxf(smx[t], smx[t + k]); }
    __syncthreads();
  }
  if (t == 0) { pmin[b * 128 + c] = smn[0]; pmax[b * 128 + c] = smx[0]; }
}
__global__ void k_mm2(float* mm, const float* __restrict__ pmin, const float* __restrict__ pmax)
{
  __shared__ float smn[128], smx[128];
  int b = blockIdx.x, t = threadIdx.x;
  smn[t] = pmin[b * 128 + t]; smx[t] = pmax[b * 128 + t]; __syncthreads();
  for (int k = 64; k > 0; k >>= 1) {
    if (t < k) { smn[t] = fminf(smn[t], smn[t + k]); smx[t] = fmaxf(smx[t], smx[t + k]); }
    __syncthreads();
  }
  if (t == 0) { mm[2 * b] = smn[0]; mm[2 * b + 1] = smx[0]; }
}

// ---------------- total, median (histogram select), edges ----------------
__global__ void k_total(float* tot, const float* __restrict__ de, const float* __restrict__ dh,
                        const float* __restrict__ mmE, const float* __restrict__ mmH,
                        const float* __restrict__ ap)
{
  size_t idx = (size_t)blockIdx.x * 256 + threadIdx.x;
  if (idx >= (size_t)KB * KN * KN) return;
  size_t b = idx / ((size_t)KN * KN);
  size_t rem = idx - b * (size_t)KN * KN;
  int i = (int)(rem / KN), j = (int)(rem % KN);
  if (i == j) { tot[idx] = 0.f; return; }
  float a = d_sigmoid(ap[0]);
  float den_e = fmaxf(mmE[2 * b + 1] - mmE[2 * b], 1e-8f);
  float den_h = fmaxf(mmH[2 * b + 1] - mmH[2 * b], 1e-8f);
  float dne = fminf(fmaxf((de[idx] - mmE[2 * b]) / den_e, 0.f), 1.f);
  float dnh = fminf(fmaxf((dh[idx] - mmH[2 * b]) / den_h, 0.f), 1.f);
  tot[idx] = (1.f - a) * dne * dne + a * dnh * dnh;
}

__global__ void k_qinit(float* qs, const float* __restrict__ mmT)
{
  int b = threadIdx.x;
  if (b >= KB) return;
  qs[b * 4] = mmT[2 * b]; qs[b * 4 + 1] = mmT[2 * b + 1] + 1e-6f;
  qs[b * 4 + 2] = 0.f; qs[b * 4 + 3] = 0.f;
}

__global__ void k_hist(unsigned int* hist, const float* __restrict__ tot, const float* __restrict__ qs)
{
  __shared__ unsigned int lh[1024];
  int b = blockIdx.y, c = blockIdx.x, t = threadIdx.x;
  for (int k = t; k < 1024; k += 256) lh[k] = 0;
  __syncthreads();
  float lo = qs[b * 4], hi = qs[b * 4 + 1];
  float inv = 1024.f / (hi - lo);
  size_t base = (size_t)b * KN * KN;
  for (int it = 0; it < 32; it++) {
    int pos = c * 8192 + it * 256 + t;
    int i = pos >> 10, j = pos & 1023;
    if (i != j) {
      float v = tot[base + pos];
      if (v >= lo && v < hi) {
        int bin = (int)((v - lo) * inv); if (bin > 1023) bin = 1023;
        atomicAdd(&lh[bin], 1u);
      }
    }
  }
  __syncthreads();
  for (int k = t; k < 1024; k += 256) if (lh[k]) atomicAdd(&hist[b * 1024 + k], lh[k]);
}

__global__ void k_qsel(float* qs, const unsigned int* __restrict__ hist, int fin)
{
  int b = blockIdx.x;
  if (threadIdx.x != 0) return;
  float lo = qs[b * 4], hi = qs[b * 4 + 1];
  float w = (hi - lo) * (1.f / 1024.f);
  unsigned long long base = (unsigned long long)qs[b * 4 + 2];
  const unsigned long long K0 = ((unsigned long long)KN * (KN - 1) - 1) / 2;  // rank 523775
  unsigned long long cum = base; int tb = 1023;
  for (int k = 0; k < 1024; k++) {
    unsigned int c = hist[b * 1024 + k];
    if (cum + c > K0) { tb = k; break; }
    cum += c;
  }
  qs[b * 4] = lo + tb * w; qs[b * 4 + 1] = lo + (tb + 1) * w; qs[b * 4 + 2] = (float)cum;
  if (fin) qs[b * 4 + 3] = lo + (tb + 0.5f) * w;
}

__global__ void k_edges(float* E, const float* __restrict__ tot, const float* __restrict__ mmT,
                        const float* __restrict__ qs, const float* __restrict__ tp)
{
  size_t idx = (size_t)blockIdx.x * 256 + threadIdx.x;
  if (idx >= (size_t)KB * KN * KN) return;
  size_t b = idx / ((size_t)KN * KN);
  size_t rem = idx - b * (size_t)KN * KN;
  int i = (int)(rem / KN), j = (int)(rem % KN);
  float temp = d_softplus(tp[0]) + 1e-6f;
  float th = qs[b * 4 + 3];
  float mx = d_sigmoid(temp * (th - mmT[2 * b]));       // sigmoid monotone: max at min(total)
  float mn = d_sigmoid(temp * (th - mmT[2 * b + 1]));
  float sig = d_sigmoid(temp * (th - tot[idx]));
  E[idx] = (i == j) ? 1.f : (sig - mn) / (mx - mn + 1e-8f);
}

// ---------------- host orchestration ----------------
extern "C" void kernel_launch(void* const* d_in, const int* in_sizes, int n_in,
                              void* d_out, int out_size, void* d_ws, size_t ws_size,
                              hipStream_t stream)
{
  (void)in_sizes; (void)n_in; (void)out_size; (void)ws_size;
  const float* x      = (const float*)d_in[0];
  const float* ew     = (const float*)d_in[1];
  const float* edges0 = (const float*)d_in[2];
  const float* fus_w  = (const float*)d_in[3];
  const float* fus_b  = (const float*)d_in[4];
  const float* ln_g   = (const float*)d_in[5];
  const float* ln_b   = (const float*)d_in[6];
  const float* g1_W0  = (const float*)d_in[17];
  const float* g1_W1  = (const float*)d_in[18];
  const float* g1_b   = (const float*)d_in[19];
  const float* g2_W0  = (const float*)d_in[20];
  const float* g2_W1  = (const float*)d_in[21];
  const float* g2_b   = (const float*)d_in[22];
  const float* tp     = (const float*)d_in[23];
  const float* ap     = (const float*)d_in[24];

  const size_t NN = (size_t)KB * KN * KN;
  float* ws = (float*)d_ws;
  float* S0 = ws; float* S1 = S0 + NN; float* S2 = S1 + NN; float* S3 = S2 + NN;
  float* ptr = S3 + NN;
  auto alloc = [&](size_t n){ float* r = ptr; ptr += n; return r; };
  float* pbuf  = alloc((size_t)KB * KN * KKRW);
  float* xp    = alloc((size_t)KB * KN * KDX);
  float* gxe0  = alloc((size_t)KB * KN * KHID);
  float* gxe1  = alloc((size_t)KB * KN * KHID);
  float* Ybuf  = alloc((size_t)KB * KN * KHID);
  float* Ubuf  = alloc((size_t)KB * KN * KH1P);
  float* Vbuf  = alloc((size_t)KB * KN * KH1P);
  float* gxh0  = alloc((size_t)KB * KN * KH1P);
  float* gxh1  = alloc((size_t)KB * KN * KH1P);
  float* lxbuf = alloc((size_t)KB * KN * KH1P);
  float* wpad  = alloc((size_t)KH1P * KH1P);
  float* dinv  = alloc((size_t)KB * KN);
  float* sqe   = alloc((size_t)KB * KN);
  float* h0v   = alloc((size_t)KB * KN);
  float* lhb   = alloc(256);
  float* mmE   = alloc(2 * KB);
  float* mmH   = alloc(2 * KB);
  float* mmT   = alloc(2 * KB);
  float* pmin  = alloc((size_t)KB * 128);
  float* pmax  = alloc((size_t)KB * 128);
  float* qs    = alloc(4 * KB);
  unsigned int* hist = (unsigned int*)alloc((size_t)KB * 1024);

  dim3 rowsG(KN, KB), rows256(256);
  int gNN = (int)((NN + 255) / 256);
  int gBN = (KB * KN + 255) / 256;

  auto gemm = [&](float* C, const float* A, const float* Bm, int M, int K, int Nc,
                  long sA, long sB, long sC, int tB) {
    dim3 g(Nc / 32, M / 64, KB);
    k_gemm<<<g, dim3(128), 0, stream>>>(C, A, Bm, K, Nc, sA, sB, sC, tB);
  };
  auto minmax = [&](float* mm, const float* M) {
    k_mm1<<<dim3(128, KB), 256, 0, stream>>>(pmin, pmax, M);
    k_mm2<<<KB, 128, 0, stream>>>(mm, pmin, pmax);
  };

  // ---- RRWP: T, T^2, T^3, diag(T^4) ----
  k_rrwp_T<<<rowsG, rows256, 0, stream>>>(S0, edges0, ew);
  k_diag<<<gBN, 256, 0, stream>>>(pbuf, 0, S0);
  gemm(S1, S0, S0, KN, KN, KN, (long)KN * KN, (long)KN * KN, (long)KN * KN, 0);
  k_diag<<<gBN, 256, 0, stream>>>(pbuf, 1, S1);
  gemm(S2, S1, S0, KN, KN, KN, (long)KN * KN, (long)KN * KN, (long)KN * KN, 0);
  k_diag<<<gBN, 256, 0, stream>>>(pbuf, 2, S2);
  k_diag4<<<rowsG, rows256, 0, stream>>>(pbuf, S2, S0);

  // ---- fuse + LN ; init hyperbolic embedding (65 dims, stride 80) ----
  k_fuse_ln<<<rowsG, rows256, 0, stream>>>(xp, x, pbuf, fus_w, fus_b, ln_g, ln_b);
  k_expproj<<<rowsG, rows256, 0, stream>>>(gxh0, xp, KD1, 1, KD1P, KDX);

  // ---- two graph-rewiring layers ----
  const float* ecur = edges0;
  const float* cure = xp; int Dine = KDX;
  float* curh = gxh0;
  for (int L = 0; L < 2; ++L) {
    const float* eW0 = (const float*)d_in[L ? 10 : 7];
    const float* eW1 = (const float*)d_in[L ? 11 : 8];
    const float* eb  = (const float*)d_in[L ? 12 : 9];
    const float* hW  = (const float*)d_in[L ? 15 : 13];
    const float* hb  = (const float*)d_in[L ? 16 : 14];
    int DhIn  = L ? KH1  : KD1;    // raw input dim (65 / 129)
    int DhInP = L ? KH1P : KD1P;   // padded stride (80 / 160)
    float* nexte = L ? gxe1 : gxe0;
    float* nexth = L ? gxh0 : gxh1;

    // cheb1(gx_e)
    k_dinv<<<rowsG, rows256, 0, stream>>>(dinv, ecur);
    k_make_an<<<gNN, 256, 0, stream>>>(S0, ecur, dinv);
    gemm(Ybuf, S0, cure, KN, KN, Dine, (long)KN * KN, (long)KN * Dine, (long)KN * Dine, 0);
    gemm(Ubuf, cure, eW0, KN, Dine, KHID, (long)KN * Dine, 0, (long)KN * KHID, 0);
    gemm(Vbuf, Ybuf, eW1, KN, Dine, KHID, (long)KN * Dine, 0, (long)KN * KHID, 0);
    k_combine<<<(KB * KN * KHID + 255) / 256, 256, 0, stream>>>(nexte, Ubuf, Vbuf, eb, KHID, KB * KN * KHID, 0);

    // hyp_layer(gx_h)
    k_hbprep<<<1, 256, 0, stream>>>(lhb, hb, KH1);
    k_padW<<<(KH1P * KH1P + 255) / 256, 256, 0, stream>>>(wpad, hW, KH1, DhIn, KH1P, DhInP);
    k_logmap<<<rowsG, rows256, 0, stream>>>(lxbuf, curh, DhIn, DhInP);
    gemm(Ubuf, lxbuf, wpad, KN, DhInP, KH1P, (long)KN * DhInP, 0, (long)KN * KH1P, 1);
    k_expproj<<<rowsG, rows256, 0, stream>>>(nexth, Ubuf, KH1, 0, KH1P, KH1P);
    k_mobius<<<rowsG, rows256, 0, stream>>>(nexth, lhb, KH1, KH1P);
    k_logmap<<<rowsG, rows256, 0, stream>>>(lxbuf, nexth, KH1, KH1P);
    gemm(Ubuf, ecur, lxbuf, KN, KN, KH1P, (long)KN * KN, (long)KN * KH1P, (long)KN * KH1P, 0);
    k_expproj<<<rowsG, rows256, 0, stream>>>(nexth, Ubuf, KH1, 0, KH1P, KH1P);
    k_mish_rows<<<rowsG, rows256, 0, stream>>>(nexth, KH1, KH1P);

    // pairwise distances (Gram via WMMA; zero padding contributes nothing)
    gemm(S0, nexte, nexte, KN, KHID, KN, (long)KN * KHID, (long)KN * KHID, (long)KN * KN, 1);
    gemm(S1, nexth, nexth, KN, KH1P, KN, (long)KN * KH1P, (long)KN * KH1P, (long)KN * KN, 1);
    k_diagvec<<<gBN, 256, 0, stream>>>(sqe, S0);
    k_col0<<<gBN, 256, 0, stream>>>(h0v, nexth, KH1P);
    k_dist_e<<<gNN, 256, 0, stream>>>(S2, S0, sqe);
    k_dist_h<<<gNN, 256, 0, stream>>>(S3, S1, h0v);
    minmax(mmE, S2);
    minmax(mmH, S3);
    k_total<<<gNN, 256, 0, stream>>>(S0, S2, S3, mmE, mmH, ap);
    minmax(mmT, S0);

    // device-side median of off-diagonal totals (3-level histogram refine)
    k_qinit<<<1, 32, 0, stream>>>(qs, mmT);
    for (int pass = 0; pass < 3; ++pass) {
      hipMemsetAsync(hist, 0, (size_t)KB * 1024 * sizeof(unsigned int), stream);
      k_hist<<<dim3(128, KB), 256, 0, stream>>>(hist, S0, qs);
      k_qsel<<<KB, 32, 0, stream>>>(qs, hist, pass == 2);
    }
    k_edges<<<gNN, 256, 0, stream>>>(S3, S0, mmT, qs, tp);

    ecur = S3; cure = nexte; Dine = KHID; curh = nexth;
  }

  // ---- final two cheb1 layers on x_p with learned edges ----
  k_dinv<<<rowsG, rows256, 0, stream>>>(dinv, S3);
  k_make_an<<<gNN, 256, 0, stream>>>(S0, S3, dinv);
  gemm(Ybuf, S0, xp, KN, KN, KDX, (long)KN * KN, (long)KN * KDX, (long)KN * KDX, 0);
  gemm(Ubuf, xp, g1_W0, KN, KDX, KHID, (long)KN * KDX, 0, (long)KN * KHID, 0);
  gemm(Vbuf, Ybuf, g1_W1, KN, KDX, KHID, (long)KN * KDX, 0, (long)KN * KHID, 0);
  k_combine<<<(KB * KN * KHID + 255) / 256, 256, 0, stream>>>(gxe0, Ubuf, Vbuf, g1_b, KHID, KB * KN * KHID, 1);
  gemm(Ybuf, S0, gxe0, KN, KN, KHID, (long)KN * KN, (long)KN * KHID, (long)KN * KHID, 0);
  gemm(Ubuf, gxe0, g2_W0, KN, KHID, KDX, (long)KN * KHID, 0, (long)KN * KDX, 0);
  gemm(Vbuf, Ybuf, g2_W1, KN, KHID, KDX, (long)KN * KHID, 0, (long)KN * KDX, 0);
  k_combine<<<(KB * KN * KDX + 255) / 256, 256, 0, stream>>>((float*)d_out, Ubuf, Vbuf, g2_b, KDX, KB * KN * KDX, 0);
}